// SAGE_90013924590234
// MI455X (gfx1250) — compile-verified
//
#include <hip/hip_runtime.h>
#include <hip/hip_bf16.h>
#include <stdint.h>

// GraphSAGE 2-layer forward, MI455X (gfx1250, wave32).
// - Edge phase: coalesced float4 gather + global_atomic_add_f32 (accumulators
//   fit in 192MB L2; ~1.5GB total traffic -> ~65us at 23.3 TB/s, the roofline).
// - GEMM phase (~26 GFLOP): V_WMMA_F32_16X16X4_F32, full f32 precision.
//   B pre-packed into fragment order -> inner loop = ds_load_b64 +
//   global_load_b128 + 2x wmma, fully unrollable.

typedef __attribute__((ext_vector_type(2))) float v2f;
typedef __attribute__((ext_vector_type(8))) float v8f;

// ---------------- zero fill (float4 grid-stride) ----------------
__global__ void sage_zero_kernel(float4* __restrict__ p, int n4) {
    int i = blockIdx.x * blockDim.x + threadIdx.x;
    int stride = gridDim.x * blockDim.x;
    float4 z = {0.f, 0.f, 0.f, 0.f};
    for (; i < n4; i += stride) p[i] = z;
}

// ---------------- edge scatter: one wave (32 lanes) per edge ----------------
template <int D, bool ADDCNT>
__global__ void sage_scatter_kernel(const float* __restrict__ feat,
                                    const long long* __restrict__ src,
                                    const long long* __restrict__ dst,
                                    float* __restrict__ agg,
                                    float* __restrict__ cnt,
                                    int E) {
    int warp = (blockIdx.x * blockDim.x + threadIdx.x) >> 5;
    int lane = threadIdx.x & 31;
    if (warp >= E) return;
    long long s = src[warp];
    long long t = dst[warp];
    const float4* fin = (const float4*)(feat + (size_t)s * D);
    float* aout = agg + (size_t)t * D;
#pragma unroll
    for (int c = 0; c < D / 128; ++c) {         // 1 (D=128) or 2 (D=256) iters
        float4 v = fin[c * 32 + lane];          // coalesced 512B per wave
        float* o = aout + (c * 32 + lane) * 4;
        atomicAdd(o + 0, v.x);
        atomicAdd(o + 1, v.y);
        atomicAdd(o + 2, v.z);
        atomicAdd(o + 3, v.w);
    }
    if (ADDCNT && lane == 0) atomicAdd(cnt + t, 1.0f);
}

// ------------- pack B = [Wl;Wr] into WMMA fragment order -------------
// float4 per (kstep, wavepair, lane): (Wt0[kk][n0], Wt0[kk+1][n0],
//                                      Wt1[kk][n1], Wt1[kk+1][n1])
// so the GEMM inner loop does one coalesced global_load_b128 per k-step.
template <int D, int NOUT>
__global__ void sage_packB_kernel(const float* __restrict__ Wl,
                                  const float* __restrict__ Wr,
                                  float4* __restrict__ Bp4) {
    constexpr int K = 2 * D;
    constexpr int NW = NOUT / 32;               // wave-pairs of col tiles
    int idx = blockIdx.x * blockDim.x + threadIdx.x;
    if (idx >= (K / 4) * NW * 32) return;
    int lane = idx & 31;
    int rest = idx >> 5;
    int w  = rest % NW;
    int ks = rest / NW;
    int kk = ks * 4 + ((lane >> 4) << 1);
    int n0 = w * 32 + (lane & 15);
    int n1 = n0 + 16;
    auto fetch = [&](int k, int n) -> float {
        return (k < D) ? Wl[(size_t)k * NOUT + n] : Wr[(size_t)(k - D) * NOUT + n];
    };
    float4 v;
    v.x = fetch(kk, n0);
    v.y = fetch(kk + 1, n0);
    v.z = fetch(kk, n1);
    v.w = fetch(kk + 1, n1);
    Bp4[idx] = v;
}

// deterministic per-element keep mask (stand-in for jax threefry dropout;
// same keep-rate 0.8 and 1/0.8 scaling, not bit-identical RNG)
__device__ __forceinline__ float sage_dropout(float v, unsigned idx) {
    unsigned h = idx * 2654435761u;
    h ^= h >> 16; h *= 2246822519u;
    h ^= h >> 13; h *= 3266489917u;
    h ^= h >> 16;
    return (h < 3435973837u) ? v * 1.25f : 0.0f;
}

// ---------------- fused SAGE GEMM ----------------
// out[row0:row0+16, :] = [agg/cnt | self] @ [Wl;Wr] + bias   (K = 2*D)
// Block = NOUT/32 waves; wave w owns cols [32w, 32w+32) as two 16x16 tiles
// sharing one A fragment (two independent WMMA accumulation chains).
template <int D, int NOUT, bool DROP>
__global__ __launch_bounds__(NOUT) void sage_gemm_kernel(
        const float* __restrict__ aggBuf,     // un-normalized neighbor sums
        const float* __restrict__ cnt,        // degrees (float)
        const float* __restrict__ selfBuf,
        const float4* __restrict__ Bp4,       // packed [Wl;Wr] fragments
        const float* __restrict__ bias,
        float* __restrict__ out) {
    constexpr int K   = 2 * D;
    constexpr int LDA = K + 4;                // float4-aligned, bank-spread
    extern __shared__ float Asm[];            // 16 x LDA
    const int row0 = blockIdx.x * 16;
    const int tid  = threadIdx.x;

    // cooperative A staging, mean fused: A[m][k<D] = agg/deg, A[m][k>=D] = self
#pragma unroll
    for (int it = 0; it < (16 * K / 4) / NOUT; ++it) {
        int idx4 = it * NOUT + tid;           // float4 index
        int m  = idx4 / (K / 4);
        int k  = (idx4 - m * (K / 4)) * 4;
        int r  = row0 + m;
        float4 v;
        if (k < D) {
            v = *(const float4*)(aggBuf + (size_t)r * D + k);
            float rinv = 1.0f / fmaxf(cnt[r], 1.0f);
            v.x *= rinv; v.y *= rinv; v.z *= rinv; v.w *= rinv;
        } else {
            v = *(const float4*)(selfBuf + (size_t)r * D + (k - D));
        }
        *(float4*)(Asm + m * LDA + k) = v;    // aligned: LDA%4==0, k%4==0
    }
    __syncthreads();

    const int lane = tid & 31;
    const int wave = tid >> 5;
    const int mrow = lane & 15;
    const int koff = (lane >> 4) << 1;        // 0 or 2 (ISA A/B frag layout)
    const int n0   = wave * 32 + (lane & 15);
    const int n1   = n0 + 16;

    const float4* bptr = Bp4 + (size_t)wave * 32 + lane;   // stride NOUT float4/kstep
    const float*  aptr = Asm + mrow * LDA + koff;

    v8f acc0 = {};
    v8f acc1 = {};
#pragma unroll 8
    for (int ks = 0; ks < K / 4; ++ks) {
        v2f a = *(const v2f*)(aptr + ks * 4);              // ds_load_b64
        float4 b4 = bptr[(size_t)ks * (NOUT / 32) * 32];   // global_load_b128
        v2f b0; b0.x = b4.x; b0.y = b4.y;
        v2f b1; b1.x = b4.z; b1.y = b4.w;
        acc0 = __builtin_amdgcn_wmma_f32_16x16x4_f32(
            false, a, false, b0, (short)0, acc0, false, false);
        acc1 = __builtin_amdgcn_wmma_f32_16x16x4_f32(
            false, a, false, b1, (short)0, acc1, false, false);
    }

    const float bs0 = bias[n0];
    const float bs1 = bias[n1];
    const int rbase = row0 + ((lane >> 4) << 3);  // C/D: VGPR r -> row r + 8*half
#pragma unroll
    for (int r = 0; r < 8; ++r) {
        int row = rbase + r;
        float v0 = acc0[r] + bs0;
        float v1 = acc1[r] + bs1;
        if (DROP) {
            v0 = sage_dropout(fmaxf(v0, 0.0f), (unsigned)(row * NOUT + n0));
            v1 = sage_dropout(fmaxf(v1, 0.0f), (unsigned)(row * NOUT + n1));
        }
        out[(size_t)row * NOUT + n0] = v0;
        out[(size_t)row * NOUT + n1] = v1;
    }
}

extern "C" void kernel_launch(void* const* d_in, const int* in_sizes, int n_in,
                              void* d_out, int out_size, void* d_ws, size_t ws_size,
                              hipStream_t stream) {
    const float*     x   = (const float*)d_in[0];
    const long long* ei  = (const long long*)d_in[1];   // int64 per reference
    const float*     W1l = (const float*)d_in[2];
    const float*     b1l = (const float*)d_in[3];
    const float*     W1r = (const float*)d_in[4];
    const float*     W2l = (const float*)d_in[5];
    const float*     b2l = (const float*)d_in[6];
    const float*     W2r = (const float*)d_in[7];
    float*           out = (float*)d_out;

    const int N = in_sizes[0] / 128;
    const int E = in_sizes[1] / 2;
    const long long* src = ei;
    const long long* dst = ei + E;

    // workspace: cnt[N] | agg[N*256] | h1[N*256] | Bp1[16384 f4] | Bp2[16384 f4]
    float*  cnt = (float*)d_ws;
    float*  agg = cnt + (size_t)N;
    float*  h1  = agg + (size_t)N * 256;
    float4* Bp1 = (float4*)(h1 + (size_t)N * 256);
    float4* Bp2 = Bp1 + 16384;

    const int ZT = 256;
    const int rowBlocks = N / 16;                        // 6250

    // ---- weight packing (tiny: 2 x 256KB) ----
    sage_packB_kernel<128, 256><<<dim3(64), dim3(256), 0, stream>>>(W1l, W1r, Bp1);
    sage_packB_kernel<256, 128><<<dim3(64), dim3(256), 0, stream>>>(W2l, W2r, Bp2);

    // ---- Layer 1 ----
    {   // zero cnt + agg(128-wide region): N*(1+128) floats contiguous
        int n4 = (int)(((size_t)N * 129) / 4);
        sage_zero_kernel<<<dim3((n4 + ZT - 1) / ZT), dim3(ZT), 0, stream>>>((float4*)cnt, n4);
    }
    sage_scatter_kernel<128, true><<<dim3((E + 7) / 8), dim3(256), 0, stream>>>(
        x, src, dst, agg, cnt, E);
    {   // h1 = drop(relu([agg/cnt | x] @ [W1l;W1r] + b1))   K=256, NOUT=256
        size_t shmem = (size_t)16 * (2 * 128 + 4) * sizeof(float);
        sage_gemm_kernel<128, 256, true><<<dim3(rowBlocks), dim3(256), shmem, stream>>>(
            agg, cnt, x, Bp1, b1l, h1);
    }

    // ---- Layer 2 ---- (cnt reused: degrees identical)
    {   // zero agg (256-wide)
        int n4 = (int)(((size_t)N * 256) / 4);
        sage_zero_kernel<<<dim3((n4 + ZT - 1) / ZT), dim3(ZT), 0, stream>>>((float4*)agg, n4);
    }
    sage_scatter_kernel<256, false><<<dim3((E + 7) / 8), dim3(256), 0, stream>>>(
        h1, src, dst, agg, cnt, E);
    {   // out = [agg/cnt | h1] @ [W2l;W2r] + b2   K=512, NOUT=128
        size_t shmem = (size_t)16 * (2 * 256 + 4) * sizeof(float);
        sage_gemm_kernel<256, 128, false><<<dim3(rowBlocks), dim3(128), shmem, stream>>>(
            agg, cnt, h1, Bp2, b2l, out);
    }
}